// GATv2Conv_77919296684490
// MI455X (gfx1250) — compile-verified
//
#include <hip/hip_runtime.h>
#include <hip/hip_bf16.h>

// GATv2: N=1024, D_IN=256, H=8, F=64, HF=512
#define GAT_N   1024
#define GAT_D   256
#define GAT_F   64
#define GAT_H   8
#define GAT_HF  512

typedef float v2f __attribute__((ext_vector_type(2)));
typedef float v8f __attribute__((ext_vector_type(8)));
typedef unsigned int v4u __attribute__((ext_vector_type(4)));
typedef int v4i __attribute__((ext_vector_type(4)));
typedef int v8i __attribute__((ext_vector_type(8)));

// ---- CDNA5 Tensor Data Mover staging (guarded; falls back to plain copy) ----
#if defined(__has_builtin)
#  if __has_builtin(__builtin_amdgcn_tensor_load_to_lds) && \
      __has_builtin(__builtin_amdgcn_s_wait_tensorcnt)
#    define GAT_HAVE_TDM 1
#  endif
#endif
#ifndef GAT_HAVE_TDM
#  define GAT_HAVE_TDM 0
#endif
// amdgpu-toolchain (therock-10.0 headers) declares the 6-arg builtin; ROCm 7.2
// declares the 5-arg form. The TDM header ships only with the former.
#if defined(__has_include)
#  if __has_include(<hip/amd_detail/amd_gfx1250_TDM.h>)
#    define GAT_TDM_6ARG 1
#  endif
#endif
#ifndef GAT_TDM_6ARG
#  define GAT_TDM_6ARG 0
#endif

#if GAT_HAVE_TDM
typedef __attribute__((address_space(3))) const void lds_cv;

__device__ __forceinline__ unsigned gat_lds_off(const void* p) {
    // ptrtoint of an LDS-addrspace pointer yields the 32-bit LDS byte offset
    return (unsigned)(uintptr_t)(lds_cv*)p;
}

// 2D f32 tile load via TDM: tile_d0 = contiguous run (elements),
// tile_d1 = rows, stride_d0 = row pitch in elements. D# per ISA 8.3/8.4.
__device__ __forceinline__ void gat_tdm_load_2d(const float* gptr, float* lptr,
                                                unsigned tile_d0, unsigned tile_d1,
                                                unsigned stride_d0)
{
    const unsigned long long ga = (unsigned long long)(uintptr_t)gptr;
    v4u g0;
    g0.x = 1u;                                              // count=1 (valid)
    g0.y = gat_lds_off(lptr);                               // lds_addr
    g0.z = (unsigned)ga;                                    // global_addr[31:0]
    g0.w = (unsigned)((ga >> 32) & 0x1FFFFFFu) | (2u << 30); // addr[56:32], type=2

    v8i g1;
    g1[0] = (2 << 16);                                      // data_size = 4 bytes
    g1[1] = (int)((tile_d0 & 0xFFFFu) << 16);               // tensor_dim0[15:0]
    g1[2] = (int)(((tile_d0 >> 16) & 0xFFFFu) |
                  ((tile_d1 & 0xFFFFu) << 16));             // tdim0 hi | tensor_dim1 lo
    g1[3] = (int)(((tile_d1 >> 16) & 0xFFFFu) |
                  ((tile_d0 & 0xFFFFu) << 16));             // tdim1 hi | tile_dim0
    g1[4] = (int)(tile_d1 & 0xFFFFu);                       // tile_dim1, tile_dim2=0
    g1[5] = (int)stride_d0;                                 // tensor_dim0_stride[31:0]
    g1[6] = 0;
    g1[7] = 0;

#if GAT_TDM_6ARG
    v4i z4 = {0, 0, 0, 0};
    v8i z8 = {0, 0, 0, 0, 0, 0, 0, 0};
    __builtin_amdgcn_tensor_load_to_lds(g0, g1, z4, z4, z8, 0);
#else
    v4i z4 = {0, 0, 0, 0};
    __builtin_amdgcn_tensor_load_to_lds(g0, g1, z4, z4, 0);
#endif
}
#endif // GAT_HAVE_TDM

// ---------------------------------------------------------------------------
// Generalized f32 WMMA GEMM: C = A @ B, one 16x16 C tile per wave (8 waves /
// block). V_WMMA_F32_16X16X4_F32 keeps exact f32 accumulation (reference is
// f32). blockIdx.z selects a batch slice via element strides (per-head
// attn @ h_r).
// ---------------------------------------------------------------------------
__global__ void __launch_bounds__(256) wmma_gemm_f32(
    const float* __restrict__ A, int lda, size_t saz,
    const float* __restrict__ B, int ldb, size_t sbz,
    float* __restrict__ C, int ldc, size_t scz,
    int ntilesN, int K)
{
    A += (size_t)blockIdx.z * saz;
    B += (size_t)blockIdx.z * sbz;
    C += (size_t)blockIdx.z * scz;

    const int lane = threadIdx.x & 31;
    const int wave = threadIdx.x >> 5;
    const int tile = blockIdx.x * 8 + wave;
    const int tm   = tile / ntilesN;
    const int tn   = tile % ntilesN;
    const int l15  = lane & 15;
    const int kb   = (lane >> 4) << 1;      // K pair: 0 (lanes 0-15) / 2 (16-31)

    const float* arow = A + (size_t)((tm << 4) + l15) * lda + kb;
    const float* bcol = B + (size_t)kb * ldb + (tn << 4) + l15;

    v8f acc = {};
#pragma unroll 4
    for (int k0 = 0; k0 < K; k0 += 4) {
        v2f a;
        a.x = arow[k0];
        a.y = arow[k0 + 1];
        v2f b;
        b.x = bcol[(size_t)k0 * ldb];
        b.y = bcol[(size_t)(k0 + 1) * ldb];
        acc = __builtin_amdgcn_wmma_f32_16x16x4_f32(
            false, a, false, b, (short)0, acc, false, false);
    }

    float* crow = C + (size_t)((tm << 4) + ((lane >> 4) << 3)) * ldc + (tn << 4) + l15;
#pragma unroll
    for (int r = 0; r < 8; ++r)
        crow[(size_t)r * ldc] = acc[r];
}

// ---------------------------------------------------------------------------
// scores[h][i][j] = sum_f LeakyReLU(h_l[i,h,f] + h_r[j,h,f], 0.2) * Wa[h,f],
// masked with adj (else -1e9).
// Block = (16-row i-tile, h): each HR row read once serves 16 output rows,
// cutting L2 traffic 16x vs one-row blocks. hl tile + Wa staged into LDS via
// the Tensor Data Mover; all LDS reads are wave-uniform broadcasts.
// ---------------------------------------------------------------------------
__global__ void __launch_bounds__(256) gat_scores(
    const float* __restrict__ HL, const float* __restrict__ HR,
    const int* __restrict__ adj, const float* __restrict__ Wa,
    float* __restrict__ S)
{
    const int it = blockIdx.x;              // i-tile: 0..63
    const int h  = blockIdx.y;
    const int t  = threadIdx.x;
    const int i0 = it << 4;

    __shared__ float hl_s[16][GAT_F];       // 4 KB
    __shared__ float wa_s[GAT_F];           // 256 B

#if GAT_HAVE_TDM
    if (t < 32) {   // wave 0 drives the TDM; EXEC is ignored by tensor ops
        gat_tdm_load_2d(HL + (size_t)i0 * GAT_HF + h * GAT_F, &hl_s[0][0],
                        /*tile_d0=*/GAT_F, /*tile_d1=*/16, /*stride_d0=*/GAT_HF);
        gat_tdm_load_2d(Wa + h * GAT_F, wa_s,
                        /*tile_d0=*/GAT_F, /*tile_d1=*/1, /*stride_d0=*/GAT_F);
        __builtin_amdgcn_s_wait_tensorcnt(0);
    }
    __syncthreads();
#else
    {   // fallback: 256 threads x one float4 each
        const int row = t >> 4;
        const int c4  = (t & 15) << 2;
        *(float4*)&hl_s[row][c4] =
            *(const float4*)(HL + (size_t)(i0 + row) * GAT_HF + h * GAT_F + c4);
        if (t < 16)
            *(float4*)&wa_s[t << 2] = *(const float4*)(Wa + h * GAT_F + (t << 2));
    }
    __syncthreads();
#endif

#pragma unroll
    for (int jj = 0; jj < 4; ++jj) {
        const int j = jj * 256 + t;
        const float* hr = HR + (size_t)j * GAT_HF + h * GAT_F;
        if (jj < 3)  // pull next j-stripe toward the WGP (global_prefetch_b8)
            __builtin_prefetch(hr + 256 * GAT_HF, 0, 0);

        float s[16];
#pragma unroll
        for (int rr = 0; rr < 16; ++rr) s[rr] = 0.0f;

#pragma unroll 4
        for (int f4 = 0; f4 < 16; ++f4) {
            const float4 r = *(const float4*)(hr + (f4 << 2));
            const float4 w = *(const float4*)(&wa_s[f4 << 2]);
#pragma unroll
            for (int rr = 0; rr < 16; ++rr) {
                const float4 hl4 = *(const float4*)(&hl_s[rr][f4 << 2]);
                float a0 = hl4.x + r.x;
                float a1 = hl4.y + r.y;
                float a2 = hl4.z + r.z;
                float a3 = hl4.w + r.w;
                s[rr] = fmaf(fmaxf(a0, 0.2f * a0), w.x, s[rr]);
                s[rr] = fmaf(fmaxf(a1, 0.2f * a1), w.y, s[rr]);
                s[rr] = fmaf(fmaxf(a2, 0.2f * a2), w.z, s[rr]);
                s[rr] = fmaf(fmaxf(a3, 0.2f * a3), w.w, s[rr]);
            }
        }

#pragma unroll
        for (int rr = 0; rr < 16; ++rr) {
            const int a = adj[(size_t)(i0 + rr) * GAT_N + j];
            S[((size_t)h * GAT_N + (i0 + rr)) * GAT_N + j] = (a > 0) ? s[rr] : -1e9f;
        }
    }
}

// ---------------------------------------------------------------------------
// In-place softmax over j for one (i, h) score row of length N=1024.
// ---------------------------------------------------------------------------
__global__ void __launch_bounds__(256) gat_softmax(float* __restrict__ S)
{
    const int i = blockIdx.x;
    const int h = blockIdx.y;
    const int t = threadIdx.x;
    float* row = S + ((size_t)h * GAT_N + i) * GAT_N;

    __shared__ float red[256];

    const float v0 = row[t];
    const float v1 = row[t + 256];
    const float v2 = row[t + 512];
    const float v3 = row[t + 768];

    red[t] = fmaxf(fmaxf(v0, v1), fmaxf(v2, v3));
    __syncthreads();
#pragma unroll
    for (int o = 128; o > 0; o >>= 1) {
        if (t < o) red[t] = fmaxf(red[t], red[t + o]);
        __syncthreads();
    }
    const float mx = red[0];
    __syncthreads();

    const float e0 = __expf(v0 - mx);
    const float e1 = __expf(v1 - mx);
    const float e2 = __expf(v2 - mx);
    const float e3 = __expf(v3 - mx);

    red[t] = e0 + e1 + e2 + e3;
    __syncthreads();
#pragma unroll
    for (int o = 128; o > 0; o >>= 1) {
        if (t < o) red[t] += red[t + o];
        __syncthreads();
    }
    const float inv = 1.0f / red[0];

    row[t]       = e0 * inv;
    row[t + 256] = e1 * inv;
    row[t + 512] = e2 * inv;
    row[t + 768] = e3 * inv;
}

// ---------------------------------------------------------------------------
// Launch: inputs are (x, adj, Wl, Wr, Wa) per setup_inputs order.
// Scratch layout: HL (2MB) | HR (2MB) | S (32MB)  => needs 36MB of d_ws.
// ---------------------------------------------------------------------------
extern "C" void kernel_launch(void* const* d_in, const int* in_sizes, int n_in,
                              void* d_out, int out_size, void* d_ws, size_t ws_size,
                              hipStream_t stream) {
    (void)in_sizes; (void)n_in; (void)out_size; (void)ws_size;

    const float* x   = (const float*)d_in[0];
    const int*   adj = (const int*)  d_in[1];
    const float* Wl  = (const float*)d_in[2];
    const float* Wr  = (const float*)d_in[3];
    const float* Wa  = (const float*)d_in[4];
    float*       out = (float*)d_out;

    float* HL = (float*)d_ws;                         // 1024 x 512
    float* HR = HL + (size_t)GAT_N * GAT_HF;          // 1024 x 512
    float* S  = HR + (size_t)GAT_N * GAT_HF;          // 8 x 1024 x 1024

    // h_l = x @ Wl ; h_r = x @ Wr   (1024x256 @ 256x512, f32 WMMA)
    wmma_gemm_f32<<<dim3(256, 1, 1), 256, 0, stream>>>(
        x, GAT_D, 0, Wl, GAT_HF, 0, HL, GAT_HF, 0, GAT_HF / 16, GAT_D);
    wmma_gemm_f32<<<dim3(256, 1, 1), 256, 0, stream>>>(
        x, GAT_D, 0, Wr, GAT_HF, 0, HR, GAT_HF, 0, GAT_HF / 16, GAT_D);

    // masked attention scores -> S[h][i][j]  (block = 16-row i-tile x head)
    gat_scores<<<dim3(GAT_N / 16, GAT_H), 256, 0, stream>>>(HL, HR, adj, Wa, S);

    // row softmax over j (in place)
    gat_softmax<<<dim3(GAT_N, GAT_H), 256, 0, stream>>>(S);

    // out[:, h*64:(h+1)*64] = P_h(1024x1024) @ HR_h(1024x64)  per head h
    wmma_gemm_f32<<<dim3(32, 1, GAT_H), 256, 0, stream>>>(
        S, GAT_N, (size_t)GAT_N * GAT_N,
        HR, GAT_HF, (size_t)GAT_F,
        out, GAT_HF, (size_t)GAT_F,
        GAT_F / 16, GAT_N);
}